// ShadowRouter_47794396070044
// MI455X (gfx1250) — compile-verified
//
#include <hip/hip_runtime.h>

typedef __attribute__((ext_vector_type(2))) float v2f;
typedef __attribute__((ext_vector_type(4))) float v4f;
typedef __attribute__((ext_vector_type(8))) float v8f;

#define HID 768
#define NEXP 4
#define ROWS_PER_WAVE 16
#define WAVES_PER_BLOCK 8
#define ROWS_PER_BLOCK 128
#define NTHREADS 256

// D = A(16x4 f32) x B(4x16 f32) + C(16x16 f32), accumulated over K chunks of 4.
// A rows  = 16 consecutive token rows of x.
// B cols  = experts (only 0..3 live; W padded with zero rows 4..15 in LDS).
//
// Lane mapping (wave32), per ISA 7.12.2:
//   A 16x4:  lane L (m=L&15, h=L>>4): VGPR0=A[m][h*2], VGPR1=A[m][h*2+1]  -> v2f @ x[row m] + h*2
//   B 4x16:  lane L (n=L&15, h=L>>4): VGPR0=B[h*2][n], VGPR1=B[h*2+1][n]  -> v2f @ W[n] + h*2
//   C 16x16: VGPR j, lanes 0-15: M=j, N=lane; lanes 16-31: M=j+8, N=lane-16
__global__ __launch_bounds__(NTHREADS) void shadow_router_wmma_kernel(
    const float* __restrict__ x, const float* __restrict__ W,
    const float* __restrict__ u, const int* __restrict__ step_p,
    float* __restrict__ out, int nrows) {
  __shared__ float sW[16 * HID];                 // W zero-padded to 16 "experts"
  __shared__ float sC[ROWS_PER_BLOCK * NEXP];   // logits staging for transpose

  const int tid = threadIdx.x;

  // Stage W into LDS, zero-padding experts 4..15 so B-frag loads need no mask.
  for (int i = tid; i < 16 * HID; i += NTHREADS)
    sW[i] = (i < NEXP * HID) ? W[i] : 0.0f;
  __syncthreads();

  const int wave = tid >> 5;
  const int lane = tid & 31;
  const int hl   = lane >> 4;   // K-half select: 0 -> K{0,1}, 1 -> K{2,3}
  const int m    = lane & 15;   // A row within tile / B column (expert)

  const int rowbase = blockIdx.x * ROWS_PER_BLOCK + wave * ROWS_PER_WAVE;
  int grow_m = rowbase + m;
  if (grow_m >= nrows) grow_m = nrows - 1;       // clamp (reads only; tail-safe)

  const float* xrow = x + (size_t)grow_m * HID + hl * 2;
  const float* wrow = sW + m * HID + hl * 2;

  v8f c = {};
  for (int kb = 0; kb < HID; kb += 32) {         // 32 floats = one 128B line/row
    __builtin_prefetch(xrow + kb + 64, 0, 0);    // global_prefetch_b8, 2 lines ahead
#pragma unroll
    for (int k0 = 0; k0 < 32; k0 += 4) {
      v2f a = *(const v2f*)(xrow + kb + k0);     // global_load_b64
      v2f b = *(const v2f*)(wrow + kb + k0);     // ds_load_b64
      c = __builtin_amdgcn_wmma_f32_16x16x4_f32(
          /*neg_a=*/false, a, /*neg_b=*/false, b,
          /*c_mod=*/(short)0, c, /*reuse_a=*/false, /*reuse_b=*/false);
    }
  }

  // Scatter live columns (experts 0..3) of the C fragment to LDS row-major.
  if (m < NEXP) {
#pragma unroll
    for (int j = 0; j < 8; ++j) {
      int rl = wave * ROWS_PER_WAVE + (hl ? (j + 8) : j);
      sC[rl * NEXP + m] = c[j];
    }
  }
  __syncthreads();

  // One thread per row: gumbel noise, softmax over 4 experts, top-1, stores.
  if (tid < ROWS_PER_BLOCK) {
    int grow = blockIdx.x * ROWS_PER_BLOCK + tid;
    if (grow < nrows) {
      float lg[NEXP];
#pragma unroll
      for (int e = 0; e < NEXP; ++e) lg[e] = sC[tid * NEXP + e];

      v4f uu = *(const v4f*)(u + (size_t)grow * NEXP);

      const int step = *step_p;
      float temp;
      if (step >= 10000) {
        temp = 0.2f;
      } else {
        float pr = 1.0f - (float)step * 1e-4f;
        temp = 0.2f + 0.8f * pr * pr;
      }
      if (temp > 0.01f) {
        float it = 1.0f / temp;
#pragma unroll
        for (int e = 0; e < NEXP; ++e) {
          float nz = -__logf(-__logf(uu[e] + 1e-8f) + 1e-8f);
          lg[e] = (lg[e] + nz) * it;
        }
      }

      float mx = fmaxf(fmaxf(lg[0], lg[1]), fmaxf(lg[2], lg[3]));
      float ex[NEXP], s = 0.0f;
#pragma unroll
      for (int e = 0; e < NEXP; ++e) { ex[e] = __expf(lg[e] - mx); s += ex[e]; }
      float inv = 1.0f / s;

      v4f pr4, lo4;
      int bi = 0; float bp = ex[0] * inv;
#pragma unroll
      for (int e = 0; e < NEXP; ++e) {
        float p = ex[e] * inv;
        pr4[e] = p; lo4[e] = lg[e];
        if (e > 0 && p > bp) { bp = p; bi = e; }   // first-max on ties (jax top_k)
      }

      const size_t BS = (size_t)nrows;
      out[grow]      = bp;          // top_k_p [BS]
      out[BS + grow] = (float)bi;   // top_k_i [BS]
      *(v4f*)(out + 2 * BS + (size_t)grow * 4) = pr4;  // probs  [BS,4]
      *(v4f*)(out + 6 * BS + (size_t)grow * 4) = lo4;  // logits [BS,4]
    }
  }
}

extern "C" void kernel_launch(void* const* d_in, const int* in_sizes, int n_in,
                              void* d_out, int out_size, void* d_ws, size_t ws_size,
                              hipStream_t stream) {
  (void)n_in; (void)out_size; (void)d_ws; (void)ws_size;
  const float* x  = (const float*)d_in[0];
  const float* W  = (const float*)d_in[1];
  const float* u  = (const float*)d_in[2];
  const int* step = (const int*)d_in[3];
  float* out = (float*)d_out;

  const int nrows = in_sizes[0] / HID;  // B*S = 131072
  const int blocks = (nrows + ROWS_PER_BLOCK - 1) / ROWS_PER_BLOCK;
  shadow_router_wmma_kernel<<<blocks, NTHREADS, 0, stream>>>(x, W, u, step, out, nrows);
}